// DMTetGeometryFixedTopo_1726576856240
// MI455X (gfx1250) — compile-verified
//
#include <hip/hip_runtime.h>
#include <cmath>

#define TPB   256
#define TILES 4     // tets tiles (TPB each) streamed per block through the async pipeline

// Marching-tets case tables (match the reference's TRI_TABLE / NUM_TRI_TABLE).
__device__ __constant__ int c_tri_table[16][6] = {
    {-1,-1,-1,-1,-1,-1},{1,0,2,-1,-1,-1},{4,0,3,-1,-1,-1},{1,4,2,1,3,4},
    {3,1,5,-1,-1,-1},{2,3,0,2,5,3},{1,4,0,1,5,4},{4,2,5,-1,-1,-1},
    {4,5,2,-1,-1,-1},{4,1,0,4,5,1},{3,2,0,3,5,2},{1,3,5,-1,-1,-1},
    {4,1,2,4,3,1},{3,0,4,-1,-1,-1},{2,0,1,-1,-1,-1},{-1,-1,-1,-1,-1,-1}};
__device__ __constant__ int c_num_tri[16] = {0,1,1,2,1,2,2,1,1,2,2,1,2,1,1,0};

// Issue one per-lane 16B CDNA5 async copy (ASYNCcnt-tracked) of a tet record
// into this lane's LDS slot. Tail lanes are clamped to a safe in-bounds address.
__device__ __forceinline__ void async_tile_load(const int4* __restrict__ tets4,
                                                int tile, int F, int tid,
                                                unsigned lds_off)
{
    int f  = tile * TPB + tid;
    int fc = f < F ? f : (F - 1);
    unsigned long long gaddr = (unsigned long long)(size_t)(tets4 + fc);
    asm volatile("global_load_async_to_lds_b128 %0, %1, off"
                 :: "v"(lds_off), "v"(gaddr) : "memory");
}

// One thread per tetrahedron; each block streams TILES tiles of TPB tets
// through a double-buffered LDS pipeline. Each lane reads only its own LDS
// slot, so the per-wave ASYNCcnt is the only synchronization required.
__global__ __launch_bounds__(TPB) void mt_tet_kernel(
    const float* __restrict__ pos,      // (NV,3)
    const float* __restrict__ sdf,      // (NV,)
    const int*   __restrict__ tets,     // (F,4)
    float* __restrict__ out_verts,      // (6F,3)
    float* __restrict__ out_faces,      // (F,2,3)
    float* __restrict__ out_mask,       // (F,2)
    float* __restrict__ out_numtri,     // (F,)
    float* __restrict__ out_cross,      // (6F,)
    float* __restrict__ out_uvidx,      // (F,2,3)
    int F, int maxTile)
{
    __shared__ int4 s_tets[2][TPB];

    const int tid = threadIdx.x;
    const int4* tets4 = reinterpret_cast<const int4*>(tets);

    const unsigned lds_b[2] = {
        (unsigned)(size_t)(&s_tets[0][tid]),   // addr[31:0] = LDS offset
        (unsigned)(size_t)(&s_tets[1][tid])
    };

    const int tileBase = blockIdx.x * TILES;
    if (tileBase > maxTile) return;

    // Prime the pipeline: tile 0 of this block -> buffer 0.
    async_tile_load(tets4, tileBase, F, tid, lds_b[0]);

    for (int it = 0; it < TILES; ++it) {
        const int tile = tileBase + it;
        if (tile > maxTile) break;
        const int cur = it & 1;

        // Prefetch the next tile into the alternate buffer (clamped so the
        // final iteration issues a harmless duplicate; S_ENDPGM wait-idles it).
        const int ntile = (tile + 1 <= maxTile) ? (tile + 1) : tile;
        async_tile_load(tets4, ntile, F, tid, lds_b[cur ^ 1]);

        // Async loads complete in order: <=1 outstanding means the CURRENT
        // tile's data has landed while the prefetch is still in flight.
        asm volatile("s_wait_asynccnt 1" ::: "memory");

        const int f = tile * TPB + tid;
        if (f >= F) continue;

        const int4 t4 = s_tets[cur][tid];
        const int v[4] = {t4.x, t4.y, t4.z, t4.w};

        // Gather 4 SDF samples + 4 positions once; all 6 edges reuse them.
        float s[4], px[4], py[4], pz[4];
#pragma unroll
        for (int i = 0; i < 4; ++i) {
            const long long vi = v[i];
            s[i] = sdf[vi];
            const float* p = pos + 3ll * vi;
            px[i] = p[0]; py[i] = p[1]; pz[i] = p[2];
        }

        bool occ[4];
        int tetidx = 0;
#pragma unroll
        for (int i = 0; i < 4; ++i) {
            occ[i] = s[i] > 0.0f;
            tetidx |= occ[i] ? (1 << i) : 0;
        }
        const int  nt    = c_num_tri[tetidx];
        const bool valid = (tetidx != 0) && (tetidx != 15);

        out_numtri[f] = (float)nt;
        reinterpret_cast<float2*>(out_mask)[f] =
            make_float2((nt >= 1) ? 1.0f : 0.0f, (nt == 2) ? 1.0f : 0.0f);

        // BASE_TET_EDGES pairs (compile-time -> fully unrolled).
        const int E0[6] = {0, 0, 0, 1, 1, 2};
        const int E1[6] = {1, 2, 3, 2, 3, 3};

        float vbuf[18];   // 6 edges x 3 coords, contiguous per tet
        float cbuf[6];    // cross flags
#pragma unroll
        for (int e = 0; e < 6; ++e) {
            const int ia = E0[e], ib = E1[e];
            // Sort endpoints by vertex id (reference key = min*Nv + max).
            const bool sw = v[ib] < v[ia];
            const int a = sw ? ib : ia;
            const int b = sw ? ia : ib;

            const bool cr = valid && (occ[a] != occ[b]);
            const float s0 = s[a], s1 = s[b];
            const float denom = cr ? (s0 - s1) : 1.0f;
            const float w0 = -s1 / denom;
            const float w1 =  s0 / denom;

            vbuf[e * 3 + 0] = cr ? (w0 * px[a] + w1 * px[b]) : 0.0f;
            vbuf[e * 3 + 1] = cr ? (w0 * py[a] + w1 * py[b]) : 0.0f;
            vbuf[e * 3 + 2] = cr ? (w0 * pz[a] + w1 * pz[b]) : 0.0f;
            cbuf[e]         = cr ? 1.0f : 0.0f;
        }

        // Triangles + UV-atlas indices (per-slot edge indexing: idx_map[f][l]=f*6+l).
        float fbuf[6], ubuf[6];
        const int uvbase = f * 4;
#pragma unroll
        for (int tri = 0; tri < 2; ++tri) {
            const bool m = (tri == 0) ? (nt >= 1) : (nt == 2);
#pragma unroll
            for (int j = 0; j < 3; ++j) {
                int tt = c_tri_table[tetidx][tri * 3 + j];
                tt = tt < 0 ? 0 : (tt > 5 ? 5 : tt);
                fbuf[tri * 3 + j] = m ? (float)(f * 6 + tt) : -1.0f;
            }
            ubuf[tri * 3 + 0] = (float)uvbase;
            ubuf[tri * 3 + 1] = (float)(uvbase + tri + 1);
            ubuf[tri * 3 + 2] = (float)(uvbase + tri + 2);
        }

        // Wide (b64) stores: contiguous per-tet runs are 8B-aligned, so emit
        // explicit float2 stores the compiler can't form on its own here.
        float2* vv = reinterpret_cast<float2*>(out_verts + (long long)f * 18);
#pragma unroll
        for (int i = 0; i < 9; ++i) vv[i] = make_float2(vbuf[2*i], vbuf[2*i+1]);

        float2* cc = reinterpret_cast<float2*>(out_cross + (long long)f * 6);
        float2* ff = reinterpret_cast<float2*>(out_faces + (long long)f * 6);
        float2* uu = reinterpret_cast<float2*>(out_uvidx + (long long)f * 6);
#pragma unroll
        for (int i = 0; i < 3; ++i) {
            cc[i] = make_float2(cbuf[2*i], cbuf[2*i+1]);
            ff[i] = make_float2(fbuf[2*i], fbuf[2*i+1]);
            uu[i] = make_float2(ubuf[2*i], ubuf[2*i+1]);
        }
    }
}

// UV atlas grid: 4 uv rows per grid cell, (4*N*N, 2) floats. Pure store-bound.
__global__ __launch_bounds__(TPB) void mt_uv_kernel(float* __restrict__ out_uvs,
                                                    int N, long long total)
{
    const long long idx = (long long)blockIdx.x * TPB + threadIdx.x;
    if (idx >= total) return;

    const long long cell = idx >> 2;
    const int r  = (int)(idx & 3);
    const int ci = (int)(cell / N);                   // row    -> tex_y
    const int cj = (int)(cell - (long long)ci * N);   // column -> tex_x

    const float step = (1.0f - 1.0f / (float)N) / (float)(N - 1); // linspace step
    const float pad  = 0.9f / (float)N;
    const float x = (float)cj * step;
    const float y = (float)ci * step;

    // r: 0 -> (x,y), 1 -> (x+pad,y), 2 -> (x+pad,y+pad), 3 -> (x,y+pad)
    const float u = x + ((r == 1 || r == 2) ? pad : 0.0f);
    const float w = y + ((r >= 2)           ? pad : 0.0f);

    reinterpret_cast<float2*>(out_uvs)[idx] = make_float2(u, w);
}

extern "C" void kernel_launch(void* const* d_in, const int* in_sizes, int n_in,
                              void* d_out, int out_size, void* d_ws, size_t ws_size,
                              hipStream_t stream)
{
    const float* pos  = (const float*)d_in[0];
    const float* sdf  = (const float*)d_in[1];
    const int*   tets = (const int*)d_in[2];

    const long long F = in_sizes[2] / 4;
    float* out = (float*)d_out;

    // Output layout: outputs concatenated flat in reference return order.
    long long off = 0;
    float* o_verts = out + off; off += 6 * F * 3;   // verts   (6F,3)
    float* o_faces = out + off; off += F * 6;       // faces   (F,2,3)
    float* o_mask  = out + off; off += F * 2;       // mask    (F,2)
    float* o_nt    = out + off; off += F;           // num_tri (F,)
    float* o_cross = out + off; off += 6 * F;       // cross   (6F,)

    // Ngrid = ceil(sqrt((2F+1)//2)) computed exactly.
    const long long half = (2 * F + 1) / 2;
    long long g = (long long)std::sqrt((double)half);
    while (g > 1 && (g - 1) * (g - 1) >= half) --g;
    while (g * g < half) ++g;
    const int N = (int)g;

    float* o_uvs   = out + off; off += 4LL * N * N * 2;  // uvs    (4N^2,2)
    float* o_uvidx = out + off; off += F * 6;            // uv_idx (F,2,3)

    const long long nTiles  = (F + TPB - 1) / TPB;
    const int       maxTile = (int)(nTiles - 1);
    const unsigned  gblocks = (unsigned)((nTiles + TILES - 1) / TILES);
    mt_tet_kernel<<<gblocks, TPB, 0, stream>>>(pos, sdf, tets,
                                               o_verts, o_faces, o_mask, o_nt,
                                               o_cross, o_uvidx, (int)F, maxTile);

    const long long total_uv = 4LL * N * N;
    const unsigned  ublocks  = (unsigned)((total_uv + TPB - 1) / TPB);
    mt_uv_kernel<<<ublocks, TPB, 0, stream>>>(o_uvs, N, total_uv);
}